// GAT_9912784519853
// MI455X (gfx1250) — compile-verified
//
#include <hip/hip_runtime.h>
#include <float.h>
#include <math.h>

// ---------------------------------------------------------------------------
// 2-layer GATConv (PyG semantics) for MI455X / gfx1250.
//   Layer1: F_in=128 -> H=2, C=64 (concat -> 128), ReLU
//   Layer2: F_in=128 -> H=1, C=64 (mean over 1 head)
// GEMMs use V_WMMA_F32_16X16X4_F32 (exact fp32 — GEMM is ~5 GFLOP, the
// edge-scatter phase is the bottleneck at ~2 GB of L2-resident traffic).
// The weight matrix is staged into LDS once per block with the Tensor Data
// Mover (tensor_load_to_lds + s_wait_tensorcnt), then read via DS ops.
// ---------------------------------------------------------------------------

typedef __attribute__((ext_vector_type(2))) float        v2f;
typedef __attribute__((ext_vector_type(8))) float        v8f;
typedef __attribute__((ext_vector_type(4))) unsigned int v4u;
typedef __attribute__((ext_vector_type(8))) unsigned int v8u;

#define NNODES   100000
#define NEDGES   1600000
#define ETOT     (NEDGES + NNODES)   // with self loops
#define NEGSLOPE 0.2f
#define EPSV     1e-16f

// -------------------------------- utilities --------------------------------

__global__ void fill_kernel(float* __restrict__ p, float v, int n) {
    int i = blockIdx.x * blockDim.x + threadIdx.x;
    if (i < n) p[i] = v;
}

__device__ __forceinline__ float atomicMaxF(float* addr, float val) {
    unsigned int* ua = (unsigned int*)addr;
    unsigned int old = *ua;
    while (__uint_as_float(old) < val) {
        unsigned int assumed = old;
        old = atomicCAS(ua, assumed, __float_as_uint(val));
        if (old == assumed) break;
    }
    return __uint_as_float(old);
}

__device__ __forceinline__ void edge_pair(const int* __restrict__ ei, int e,
                                          int& s, int& d) {
    if (e < NEDGES) { s = ei[e]; d = ei[NEDGES + e]; }
    else            { s = e - NEDGES; d = s; }        // self loop
}

__device__ __forceinline__ float lrelu(float a) {
    return a > 0.0f ? a : NEGSLOPE * a;
}

// ------------------------ FP32 WMMA GEMM  C = A @ B ------------------------
// A: [M,K] row-major, B: [K,N] row-major, C: [M,N] row-major. K*N <= 16384.
// One wave32 per 16x16 C tile; V_WMMA_F32_16X16X4_F32, K stepped by 4.
// B is staged into LDS once per block by the TDM (one 2D descriptor:
// tile = full KxN, data_size = 4B), then fragments come from ds_load.
// Bs (an addrspace(3) global) is passed to the asm as a captured operand:
// without that escape, GlobalsAA sees a never-stored global and folds the
// LDS loads to undef (round 2). With it, plain (non-volatile!) loads stay
// and InferAddressSpaces lowers them to ds_load. (volatile pinned them to
// SYS-scope flat loads in round 4; "=m" LDS operands crash isel in round 3.)
// ISA fragment layout (CDNA5 ISA):
//   A 16x4 : lane L -> row M = L&15 ; VGPR0/1 = K = 2*(L>>4) + {0,1}
//   B 4x16 : lane L -> col N = L&15 ; VGPR0/1 = K = 2*(L>>4) + {0,1}
//   C 16x16: VGPR r -> row (r + 8*(L>>4)), col = L&15
// Grids are sized exactly (M % 16 == 0) so EXEC is all-ones for every wave.

__global__ void gemm_f32_wmma(const float* __restrict__ A,
                              const float* __restrict__ B,
                              float* __restrict__ C,
                              int M, int K, int N) {
    __shared__ float Bs[16384];               // 64 KB of the 320 KB WGP LDS

    // ---- stage B into LDS with the Tensor Data Mover (one wave issues) ----
    if (threadIdx.x < 32) {
        unsigned long long ga = (unsigned long long)(const void*)B;
        v4u g0;
        g0[0] = 1u;                                       // count=1, user mode
        g0[1] = 0u;                                       // lds_addr = 0 (Bs base)
        g0[2] = (unsigned)ga;                             // global_addr[31:0]
        g0[3] = (unsigned)((ga >> 32) & 0x01FFFFFFu)      // global_addr[56:32]
                | (2u << 30);                             // type = 2 ("image")
        v8u g1;
        g1[0] = (2u << 16);                               // data_size=4B, mask=0
        g1[1] = ((unsigned)N & 0xFFFFu) << 16;            // tensor_dim0[15:0]
        g1[2] = (((unsigned)N >> 16) & 0xFFFFu)           // tensor_dim0[31:16]
                | (((unsigned)K & 0xFFFFu) << 16);        // tensor_dim1[15:0]
        g1[3] = (((unsigned)K >> 16) & 0xFFFFu)           // tensor_dim1[31:16]
                | (((unsigned)N & 0xFFFFu) << 16);        // tile_dim0 = N
        g1[4] = ((unsigned)K & 0xFFFFu);                  // tile_dim1=K, tile_dim2=0
        g1[5] = (unsigned)N;                              // tensor_dim0_stride lo
        g1[6] = 0u;                                       // stride0 hi | stride1 lo
        g1[7] = 0u;                                       // stride1 hi (2D: unused)
        // Bs passed as operand %2 (unreferenced in the template) so its
        // address escapes into the memory-clobbering asm: the compiler must
        // then assume the TDM writes Bs.
        asm volatile("tensor_load_to_lds %0, %1"
                     :
                     : "s"(g0), "s"(g1), "v"(Bs)
                     : "memory");
        __builtin_amdgcn_s_wait_tensorcnt(0);
    }
    __syncthreads();

    const int lane = threadIdx.x & 31;
    const int wid  = (blockIdx.x * blockDim.x + threadIdx.x) >> 5;
    const int ntiles = N >> 4;
    const int mt = wid / ntiles;
    const int nt = wid % ntiles;
    if (mt * 16 >= M) return;                 // wave-uniform guard (never taken)

    const int half = lane >> 4;               // 0..1 -> K pair select
    const int rc   = lane & 15;               // row (A) / col (B,C)

    const float* Arow = A + (size_t)(mt * 16 + rc) * K;
    const int    bbase = nt * 16 + rc;        // column offset into Bs

    v8f acc = {};
#pragma unroll 8
    for (int k = 0; k < K; k += 4) {
        v2f a, b;
        a.x = Arow[k + 2 * half];
        a.y = Arow[k + 2 * half + 1];
        b.x = Bs[(k + 2 * half)     * N + bbase];   // ds_load
        b.y = Bs[(k + 2 * half + 1) * N + bbase];   // ds_load
        acc = __builtin_amdgcn_wmma_f32_16x16x4_f32(
                  false, a, false, b, (short)0, acc, false, false);
    }

    float* Crow = C + (size_t)(mt * 16 + 8 * half) * N + nt * 16 + rc;
#pragma unroll
    for (int r = 0; r < 8; ++r)
        Crow[(size_t)r * N] = acc[r];
}

// -------------------- per-node attention logits a_src/a_dst ----------------
// one thread per (node, head): a_* = sum_c h[n,hd,c] * att_*[hd,c]

__global__ void attn_coef_kernel(const float* __restrict__ h,
                                 const float* __restrict__ att_src,
                                 const float* __restrict__ att_dst,
                                 float* __restrict__ a_src,
                                 float* __restrict__ a_dst,
                                 int n, int H, int C) {
    int i = blockIdx.x * blockDim.x + threadIdx.x;   // node*H + head
    if (i >= n * H) return;
    int hd = i % H;
    const float* hr = h + (size_t)i * C;
    const float* as = att_src + hd * C;
    const float* ad = att_dst + hd * C;
    float s1 = 0.f, s2 = 0.f;
    for (int c = 0; c < C; ++c) {
        float v = hr[c];
        s1 += v * as[c];
        s2 += v * ad[c];
    }
    a_src[i] = s1;
    a_dst[i] = s2;
}

// ------------------------- edge pass 1: segment max ------------------------

__global__ void edge_max_kernel(const int* __restrict__ ei,
                                const float* __restrict__ a_src,
                                const float* __restrict__ a_dst,
                                float* __restrict__ m, int H) {
    int i = blockIdx.x * blockDim.x + threadIdx.x;   // edge*H + head
    if (i >= ETOT * H) return;
    int e = i / H, hd = i % H;
    int s, d; edge_pair(ei, e, s, d);
    float a = lrelu(a_src[s * H + hd] + a_dst[d * H + hd]);
    atomicMaxF(&m[d * H + hd], a);
}

// ---------------------- edge pass 2: segment sum of exp --------------------

__global__ void edge_sum_kernel(const int* __restrict__ ei,
                                const float* __restrict__ a_src,
                                const float* __restrict__ a_dst,
                                const float* __restrict__ m,
                                float* __restrict__ ssum, int H) {
    int i = blockIdx.x * blockDim.x + threadIdx.x;
    if (i >= ETOT * H) return;
    int e = i / H, hd = i % H;
    int s, d; edge_pair(ei, e, s, d);
    float a = lrelu(a_src[s * H + hd] + a_dst[d * H + hd]);
    atomicAdd(&ssum[d * H + hd], __expf(a - m[d * H + hd]));
}

// -------------------- edge pass 3: weighted aggregation --------------------
// one wave per edge; each lane handles H*C/32 features; atomics land in L2
// (out buffer is <= 51 MB, fully L2-resident on the 192 MB L2).

__global__ void edge_aggr_kernel(const int* __restrict__ ei,
                                 const float* __restrict__ a_src,
                                 const float* __restrict__ a_dst,
                                 const float* __restrict__ m,
                                 const float* __restrict__ ssum,
                                 const float* __restrict__ h,
                                 float* __restrict__ out,
                                 int H, int C) {
    const int lane = threadIdx.x & 31;
    const int e = (blockIdx.x * blockDim.x + threadIdx.x) >> 5;
    if (e >= ETOT) return;
    int s, d; edge_pair(ei, e, s, d);
    const int HC = H * C;
    const float* hsrc = h + (size_t)s * HC;
    __builtin_prefetch(hsrc + lane, 0, 0);           // global_prefetch_b8
    float* od = out + (size_t)d * HC;

    float coef[2];                                   // per-head softmax coeff
    for (int hd = 0; hd < H; ++hd) {
        float a = lrelu(a_src[s * H + hd] + a_dst[d * H + hd]);
        coef[hd] = __expf(a - m[d * H + hd]) / (ssum[d * H + hd] + EPSV);
    }
    for (int f = lane; f < HC; f += 32) {
        atomicAdd(&od[f], hsrc[f] * coef[f / C]);
    }
}

// -------------------------- epilogue: bias (+ReLU) -------------------------

__global__ void bias_act_kernel(float* __restrict__ out,
                                const float* __restrict__ bias,
                                int n, int F, int do_relu) {
    int i = blockIdx.x * blockDim.x + threadIdx.x;
    if (i >= n * F) return;
    float v = out[i] + bias[i % F];
    out[i] = do_relu ? fmaxf(v, 0.0f) : v;
}

// ---------------------------------------------------------------------------

extern "C" void kernel_launch(void* const* d_in, const int* in_sizes, int n_in,
                              void* d_out, int out_size, void* d_ws, size_t ws_size,
                              hipStream_t stream) {
    const float* x        = (const float*)d_in[0];   // [N,128]
    const int*   ei       = (const int*)  d_in[1];   // [2,E]
    const float* W1       = (const float*)d_in[2];   // [128,128]
    const float* att_src1 = (const float*)d_in[3];   // [2,64]
    const float* att_dst1 = (const float*)d_in[4];
    const float* bias1    = (const float*)d_in[5];   // [128]
    const float* W2       = (const float*)d_in[6];   // [128,64]
    const float* att_src2 = (const float*)d_in[7];   // [1,64]
    const float* att_dst2 = (const float*)d_in[8];
    const float* bias2    = (const float*)d_in[9];   // [64]

    // workspace layout (floats)
    float* ws = (float*)d_ws;
    size_t off = 0;
    float* h1     = ws + off; off += (size_t)NNODES * 128;  // layer1 features
    float* out1   = ws + off; off += (size_t)NNODES * 128;  // layer1 out / layer2 in
    float* h2     = ws + off; off += (size_t)NNODES * 64;   // layer2 features
    float* a_src1 = ws + off; off += (size_t)NNODES * 2;
    float* a_dst1 = ws + off; off += (size_t)NNODES * 2;
    float* m1     = ws + off; off += (size_t)NNODES * 2;
    float* s1     = ws + off; off += (size_t)NNODES * 2;
    float* a_src2 = ws + off; off += (size_t)NNODES;
    float* a_dst2 = ws + off; off += (size_t)NNODES;
    float* m2     = ws + off; off += (size_t)NNODES;
    float* s2     = ws + off; off += (size_t)NNODES;
    float* outf   = (float*)d_out;                          // [N,64]

    const int TB = 256;
    auto blk = [](long n, int tb) { return (int)((n + tb - 1) / tb); };

    // ---- init accumulators (every call: determinism under graph replay) ----
    fill_kernel<<<blk(2L * NNODES, TB), TB, 0, stream>>>(m1, -FLT_MAX, 2 * NNODES);
    fill_kernel<<<blk(2L * NNODES, TB), TB, 0, stream>>>(s1, 0.0f, 2 * NNODES);
    fill_kernel<<<blk(128L * NNODES, TB), TB, 0, stream>>>(out1, 0.0f, 128 * NNODES);
    fill_kernel<<<blk((long)NNODES, TB), TB, 0, stream>>>(m2, -FLT_MAX, NNODES);
    fill_kernel<<<blk((long)NNODES, TB), TB, 0, stream>>>(s2, 0.0f, NNODES);
    fill_kernel<<<blk(64L * NNODES, TB), TB, 0, stream>>>(outf, 0.0f, 64 * NNODES);

    // =========================== layer 1 (H=2, C=64) ========================
    // h1 = x @ W1 : tiles = 6250 x 8 = 50000 ; 8 waves/block -> 6250 blocks
    gemm_f32_wmma<<<6250, TB, 0, stream>>>(x, W1, h1, NNODES, 128, 128);
    attn_coef_kernel<<<blk(2L * NNODES, TB), TB, 0, stream>>>(
        h1, att_src1, att_dst1, a_src1, a_dst1, NNODES, 2, 64);
    edge_max_kernel<<<blk(2L * ETOT, TB), TB, 0, stream>>>(ei, a_src1, a_dst1, m1, 2);
    edge_sum_kernel<<<blk(2L * ETOT, TB), TB, 0, stream>>>(ei, a_src1, a_dst1, m1, s1, 2);
    edge_aggr_kernel<<<ETOT / 8, TB, 0, stream>>>(                // 1.7M waves
        ei, a_src1, a_dst1, m1, s1, h1, out1, 2, 64);
    bias_act_kernel<<<blk(128L * NNODES, TB), TB, 0, stream>>>(out1, bias1, NNODES, 128, 1);

    // =========================== layer 2 (H=1, C=64) ========================
    // h2 = out1 @ W2 : tiles = 6250 x 4 = 25000 -> 3125 blocks
    gemm_f32_wmma<<<3125, TB, 0, stream>>>(out1, W2, h2, NNODES, 128, 64);
    attn_coef_kernel<<<blk((long)NNODES, TB), TB, 0, stream>>>(
        h2, att_src2, att_dst2, a_src2, a_dst2, NNODES, 1, 64);
    edge_max_kernel<<<blk((long)ETOT, TB), TB, 0, stream>>>(ei, a_src2, a_dst2, m2, 1);
    edge_sum_kernel<<<blk((long)ETOT, TB), TB, 0, stream>>>(ei, a_src2, a_dst2, m2, s2, 1);
    edge_aggr_kernel<<<ETOT / 8, TB, 0, stream>>>(
        ei, a_src2, a_dst2, m2, s2, h2, outf, 1, 64);
    bias_act_kernel<<<blk(64L * NNODES, TB), TB, 0, stream>>>(outf, bias2, NNODES, 64, 0);
}